// LocalAttention_28681791603223
// MI455X (gfx1250) — compile-verified
//
#include <hip/hip_runtime.h>
#include <stdint.h>

typedef __attribute__((ext_vector_type(2))) float v2f;
typedef __attribute__((ext_vector_type(8))) float v8f;
typedef uint32_t su32x4 __attribute__((ext_vector_type(4)));
typedef uint32_t su32x8 __attribute__((ext_vector_type(8)));

namespace {
constexpr int kN    = 577;          // sequence length (1 CLS + 24*24)
constexpr int kNP   = 592;          // padded to multiple of 16 (37 tiles)
constexpr int kC    = 768;
constexpr int kH    = 12;
constexpr int kD    = 64;
constexpr int kB    = 32;
constexpr int kM    = kB * kN;      // 18464 flattened rows
constexpr int kQKV  = 3 * kC;       // 2304
constexpr int kGrid = 24;
constexpr int kRad  = 3;            // window//2
} // namespace

static __device__ __forceinline__ v8f wmma4(v2f a, v2f b, v8f c) {
  // V_WMMA_F32_16X16X4_F32 : D(16x16 f32) = A(16x4 f32) * B(4x16 f32) + C
  return __builtin_amdgcn_wmma_f32_16x16x4_f32(false, a, false, b, (short)0, c,
                                               false, false);
}

// Generic LDS address: low 32 bits of the generic pointer are the wave-relative
// LDS byte offset (ISA 10.2).
static __device__ __forceinline__ uint32_t lds_off(const void* p) {
  return (uint32_t)(uintptr_t)p;
}

// CDNA5 async memory->LDS copy, 16B per lane, tracked with ASYNCcnt.
static __device__ __forceinline__ void async_cp16(uint32_t dst_lds,
                                                  const void* g) {
  asm volatile("global_load_async_to_lds_b128 %0, %1, off"
               :: "v"(dst_lds), "v"((unsigned long long)(uintptr_t)g)
               : "memory");
}
static __device__ __forceinline__ void wait_async_0() {
  asm volatile("s_wait_asynccnt 0x0" ::: "memory");
}
static __device__ __forceinline__ void wait_ds_0() {
  asm volatile("s_wait_dscnt 0x0" ::: "memory");
}

// ---------------------------------------------------------------------------
// Tensor Data Mover: 2D tile load, global -> LDS.  D# per ISA ch.8:
//  group0: [1:0]=count=1, [63:32]=lds_addr, [120:64]=global_addr, [127:126]=2
//  group1: [17:16]=data_size(2->4B), [79:48]=tensor_dim0, [111:80]=tensor_dim1,
//          [127:112]=tile_dim0, [143:128]=tile_dim1, [207:160]=dim0_stride
// Tile written linearly to LDS (x fastest). OOB rows read as zero.
// ---------------------------------------------------------------------------
static __device__ __forceinline__ void tdm_load_2d(
    uint32_t lds_addr, const void* gaddr, uint32_t tile_w, uint32_t tile_h,
    uint32_t row_stride_elems, uint32_t tensor_w, uint32_t tensor_h) {
  const uint64_t ga = (uint64_t)(uintptr_t)gaddr;
  su32x4 g0;
  g0.x = 1u;                                                  // count=1
  g0.y = lds_addr;
  g0.z = (uint32_t)ga;
  g0.w = (uint32_t)((ga >> 32) & 0x1FFFFFFu) | (2u << 30);    // addr hi | type=2
  su32x8 g1;
  g1.s0 = 2u << 16;                                           // data_size = 4B
  g1.s1 = (tensor_w & 0xFFFFu) << 16;                         // tensor_dim0 lo
  g1.s2 = (tensor_w >> 16) | ((tensor_h & 0xFFFFu) << 16);    // dim0 hi|dim1 lo
  g1.s3 = (tensor_h >> 16) | (tile_w << 16);                  // dim1 hi|tile_dim0
  g1.s4 = tile_h & 0xFFFFu;                                   // tile_dim1
  g1.s5 = row_stride_elems;                                   // dim0_stride lo32
  g1.s6 = 0;
  g1.s7 = 0;
  asm volatile("tensor_load_to_lds %0, %1, null, null"
               :: "s"(g0), "s"(g1) : "memory");
}
static __device__ __forceinline__ void wait_tensor_le2() {
  asm volatile("s_wait_tensorcnt 0x2" ::: "memory");
}
static __device__ __forceinline__ void wait_tensor_0() {
  asm volatile("s_wait_tensorcnt 0x0" ::: "memory");
}

// ---------------------------------------------------------------------------
// Shared GEMM core: C[128x64 block] = A[128xK] * B[Kx64], K = 768.
// 4 waves, wave owns 32(M)x64(N). Double-buffered TDM staging: wave 0 issues
// one 2D tensor descriptor each for the A (128x16) and B (16x64) K-chunks.
// K loop unrolled by 2 so LDS buffers are named by constants (addrspace(3)).
// ---------------------------------------------------------------------------
template <int LDB>
static __device__ __forceinline__ void gemm_accum(
    const float* __restrict__ a_src, const float* __restrict__ b_src,
    int m0, int tid, int wave, int l16, int klo, v8f acc0[4], v8f acc1[4]) {
  __shared__ float abuf0[128 * 16];
  __shared__ float abuf1[128 * 16];
  __shared__ float bbuf0[16 * 64];
  __shared__ float bbuf1[16 * 64];
  const uint32_t aL0 = lds_off(abuf0), aL1 = lds_off(abuf1);
  const uint32_t bL0 = lds_off(bbuf0), bL1 = lds_off(bbuf1);
  constexpr int nch = kC / 16;          // 48 K-chunks (even)
  const uint32_t arows = (uint32_t)(kM - m0);   // remaining rows: OOB rows -> 0

  auto stage = [&](int k0, uint32_t aL, uint32_t bL) {
    // A chunk: 128 rows x 16 floats, row stride kC
    tdm_load_2d(aL, a_src + (size_t)m0 * kC + k0, 16, 128, kC, 0x100000u,
                arows);
    // B chunk: 16 rows x 64 floats, row stride LDB
    tdm_load_2d(bL, b_src + (size_t)k0 * LDB, 64, 16, LDB, 0x100000u, 16u);
  };

  auto compute = [&](const float* ab, const float* bb) {
#pragma unroll
    for (int kc = 0; kc < 4; ++kc) {
      v2f a0 = *(const v2f*)(ab + (wave * 32 + l16) * 16 + kc * 4 + klo);
      v2f a1 = *(const v2f*)(ab + (wave * 32 + 16 + l16) * 16 + kc * 4 + klo);
#pragma unroll
      for (int nt = 0; nt < 4; ++nt) {
        v2f bm;
        bm.x = bb[(kc * 4 + klo) * 64 + nt * 16 + l16];
        bm.y = bb[(kc * 4 + klo + 1) * 64 + nt * 16 + l16];
        acc0[nt] = wmma4(a0, bm, acc0[nt]);
        acc1[nt] = wmma4(a1, bm, acc1[nt]);
      }
    }
  };

  if (wave == 0) stage(0, aL0, bL0);
  for (int i = 0; i < nch; i += 2) {
    // ---- even chunk i, lives in buffer 0
    if (wave == 0) {
      if (i + 1 < nch) {
        stage((i + 1) * 16, aL1, bL1);
        wait_tensor_le2();              // oldest 2 descriptors == chunk i
      } else {
        wait_tensor_0();
      }
    }
    __syncthreads();                    // chunk i visible to all waves
    compute(abuf0, bbuf0);
    __syncthreads();                    // reads done before buffer reuse
    // ---- odd chunk i+1, lives in buffer 1
    if (wave == 0) {
      if (i + 2 < nch) {
        stage((i + 2) * 16, aL0, bL0);
        wait_tensor_le2();              // oldest 2 == chunk i+1
      } else {
        wait_tensor_0();
      }
    }
    __syncthreads();
    compute(abuf1, bbuf1);
    __syncthreads();
  }
}

// ---------------------------------------------------------------------------
// Kernel 1: QKV projection  x[18464x768] @ W_qkv[768x2304]
// scatter into Q/K/V [B, H, NP, D]
// ---------------------------------------------------------------------------
__global__ __launch_bounds__(128) void qkv_gemm_kernel(
    const float* __restrict__ x, const float* __restrict__ w,
    float* __restrict__ qb, float* __restrict__ kb, float* __restrict__ vb) {
  const int tid  = threadIdx.x;
  const int wave = tid >> 5;
  const int lane = tid & 31;
  const int l16  = lane & 15;
  const int klo  = (lane >> 4) << 1;
  const int rbase = (lane >> 4) << 3;
  const int m0 = blockIdx.y * 128;
  const int n0 = blockIdx.x * 64;

  const v8f vzero = {0.f, 0.f, 0.f, 0.f, 0.f, 0.f, 0.f, 0.f};
  v8f acc0[4], acc1[4];
#pragma unroll
  for (int nt = 0; nt < 4; ++nt) { acc0[nt] = vzero; acc1[nt] = vzero; }

  gemm_accum<kQKV>(x, w + n0, m0, tid, wave, l16, klo, acc0, acc1);

  const int which = n0 / kC;           // 0=Q 1=K 2=V (tile never straddles)
  const int head  = (n0 % kC) >> 6;    // 64-col tile == one head
  float* dst = (which == 0) ? qb : ((which == 1) ? kb : vb);
#pragma unroll
  for (int mt = 0; mt < 2; ++mt) {
#pragma unroll
    for (int nt = 0; nt < 4; ++nt) {
#pragma unroll
      for (int j = 0; j < 8; ++j) {
        int m = m0 + wave * 32 + mt * 16 + rbase + j;
        if (m < kM) {
          int bidx = m / kN;
          int n    = m - bidx * kN;
          float val = mt ? acc1[nt][j] : acc0[nt][j];
          dst[(((size_t)bidx * kH + head) * kNP + n) * kD + nt * 16 + l16] =
              val;
        }
      }
    }
  }
}

// ---------------------------------------------------------------------------
// Kernel 2: masked flash attention. One wave per (b, h, 16-query tile).
// ---------------------------------------------------------------------------
__global__ __launch_bounds__(32) void attn_kernel(
    const float* __restrict__ qb, const float* __restrict__ kb,
    const float* __restrict__ vb, float* __restrict__ ob) {
  __shared__ float kbuf[16 * 64];
  __shared__ float vbuf[16 * 64];
  __shared__ float pbuf[16 * 16];

  const int lane  = threadIdx.x & 31;
  const int l16   = lane & 15;
  const int klo   = (lane >> 4) << 1;
  const int rbase = (lane >> 4) << 3;
  const int qt    = blockIdx.x;
  const int head  = blockIdx.y;
  const int b     = blockIdx.z;
  const int bh    = b * kH + head;

  // Q tile preloaded in WMMA A-layout
  v2f qa[16];
  const float* qrow = qb + ((size_t)bh * kNP + qt * 16 + l16) * kD;
#pragma unroll
  for (int kc = 0; kc < 16; ++kc) qa[kc] = *(const v2f*)(qrow + kc * 4 + klo);

  const v8f vzero = {0.f, 0.f, 0.f, 0.f, 0.f, 0.f, 0.f, 0.f};
  v8f oacc[4];
#pragma unroll
  for (int dt = 0; dt < 4; ++dt) oacc[dt] = vzero;
  float mrow[8], lsum[8];
#pragma unroll
  for (int j = 0; j < 8; ++j) { mrow[j] = -1e30f; lsum[j] = 0.f; }

  const uint32_t kL = lds_off(kbuf);
  const uint32_t vL = lds_off(vbuf);

  for (int kt = 0; kt < kNP / 16; ++kt) {
    // ---- tile pruning by patch-row distance (CLS tiles always kept)
    bool need = (qt == 0) || (kt == 0);
    if (!need) {
      int qlo = qt * 16, qhi = min(qlo + 15, kN - 1);
      int kl2 = kt * 16, khi = min(kl2 + 15, kN - 1);
      int rqlo = (qlo - 1) / kGrid, rqhi = (qhi - 1) / kGrid;
      int rklo = (kl2 - 1) / kGrid, rkhi = (khi - 1) / kGrid;
      need = !(rklo > rqhi + kRad || rkhi < rqlo - kRad);
    }
    if (!need) continue;

    // ---- async-stage K/V tiles (16x64 each) into LDS
    wait_ds_0();   // async LDS writes are unordered vs outstanding DS reads
    const float* ksrc = kb + ((size_t)bh * kNP + kt * 16) * kD;
    const float* vsrc = vb + ((size_t)bh * kNP + kt * 16) * kD;
#pragma unroll
    for (int i = 0; i < 8; ++i) {
      int e = lane + i * 32;            // float4 index
      async_cp16(kL + e * 16, ksrc + e * 4);
      async_cp16(vL + e * 16, vsrc + e * 4);
    }
    wait_async_0();

    // ---- S = Q * K^T (16 WMMAs over d=64)
    v8f s = vzero;
#pragma unroll
    for (int kc = 0; kc < 16; ++kc) {
      v2f bm = *(const v2f*)(kbuf + l16 * kD + kc * 4 + klo);
      s = wmma4(qa[kc], bm, s);
    }

    // ---- scale + analytic local-window mask + online softmax update
    const int kcol = kt * 16 + l16;
    float corr[8], p[8];
#pragma unroll
    for (int j = 0; j < 8; ++j) {
      int qg = qt * 16 + rbase + j;
      float sv = s[j] * 0.125f;         // 1/sqrt(64)
      bool ok;
      if (qg >= kN || kcol >= kN) ok = false;
      else if (qg == 0 || kcol == 0) ok = true;
      else {
        int iq = qg - 1, ik = kcol - 1;
        int rq = iq / kGrid, cq = iq - rq * kGrid;
        int rk = ik / kGrid, ck = ik - rk * kGrid;
        int dr = rq > rk ? rq - rk : rk - rq;
        int dc = cq > ck ? cq - ck : ck - cq;
        ok = (dr <= kRad) && (dc <= kRad);
      }
      if (!ok) sv = -1e30f;
      float rmax = sv;
#pragma unroll
      for (int msk = 1; msk < 16; msk <<= 1)
        rmax = fmaxf(rmax, __shfl_xor(rmax, msk, 32));
      float mnew = fmaxf(mrow[j], rmax);
      float c  = __expf(mrow[j] - mnew);
      float pv = __expf(sv - mnew);
      float rs = pv;
#pragma unroll
      for (int msk = 1; msk < 16; msk <<= 1) rs += __shfl_xor(rs, msk, 32);
      mrow[j] = mnew;
      corr[j] = c;
      lsum[j] = lsum[j] * c + rs;
      p[j] = pv;
    }
#pragma unroll
    for (int dt = 0; dt < 4; ++dt)
#pragma unroll
      for (int j = 0; j < 8; ++j) oacc[dt][j] *= corr[j];

    // ---- bounce P through LDS into A-layout
#pragma unroll
    for (int j = 0; j < 8; ++j) pbuf[(rbase + j) * 16 + l16] = p[j];
    __syncthreads();

    // ---- O += P * V
#pragma unroll
    for (int kc = 0; kc < 4; ++kc) {
      v2f a = *(const v2f*)(pbuf + l16 * 16 + kc * 4 + klo);
#pragma unroll
      for (int dt = 0; dt < 4; ++dt) {
        v2f bm;
        bm.x = vbuf[(kc * 4 + klo) * kD + dt * 16 + l16];
        bm.y = vbuf[(kc * 4 + klo + 1) * kD + dt * 16 + l16];
        oacc[dt] = wmma4(a, bm, oacc[dt]);
      }
    }
  }

  // ---- normalize and store into [B, N, C] for the out-projection
#pragma unroll
  for (int j = 0; j < 8; ++j) {
    int qg = qt * 16 + rbase + j;
    if (qg >= kN) continue;
    float inv = 1.0f / lsum[j];
#pragma unroll
    for (int dt = 0; dt < 4; ++dt)
      ob[((size_t)(b * kN + qg)) * kC + head * kD + dt * 16 + l16] =
          oacc[dt][j] * inv;
  }
}

// ---------------------------------------------------------------------------
// Kernel 3: out projection  A[18464x768] @ W_proj[768x768] + bias
// ---------------------------------------------------------------------------
__global__ __launch_bounds__(128) void proj_gemm_kernel(
    const float* __restrict__ a_in, const float* __restrict__ w,
    const float* __restrict__ bias, float* __restrict__ out) {
  const int tid  = threadIdx.x;
  const int wave = tid >> 5;
  const int lane = tid & 31;
  const int l16  = lane & 15;
  const int klo  = (lane >> 4) << 1;
  const int rbase = (lane >> 4) << 3;
  const int m0 = blockIdx.y * 128;
  const int n0 = blockIdx.x * 64;

  const v8f vzero = {0.f, 0.f, 0.f, 0.f, 0.f, 0.f, 0.f, 0.f};
  v8f acc0[4], acc1[4];
#pragma unroll
  for (int nt = 0; nt < 4; ++nt) { acc0[nt] = vzero; acc1[nt] = vzero; }

  gemm_accum<kC>(a_in, w + n0, m0, tid, wave, l16, klo, acc0, acc1);

#pragma unroll
  for (int nt = 0; nt < 4; ++nt) {
    float bv = bias[n0 + nt * 16 + l16];
#pragma unroll
    for (int mt = 0; mt < 2; ++mt) {
#pragma unroll
      for (int j = 0; j < 8; ++j) {
        int m = m0 + wave * 32 + mt * 16 + rbase + j;
        if (m < kM) {
          float val = mt ? acc1[nt][j] : acc0[nt][j];
          out[(size_t)m * kC + n0 + nt * 16 + l16] = val + bv;
        }
      }
    }
  }
}

// ---------------------------------------------------------------------------
extern "C" void kernel_launch(void* const* d_in, const int* in_sizes, int n_in,
                              void* d_out, int out_size, void* d_ws,
                              size_t ws_size, hipStream_t stream) {
  (void)in_sizes; (void)n_in; (void)out_size; (void)ws_size;
  const float* x      = (const float*)d_in[0];
  const float* w_qkv  = (const float*)d_in[1];
  const float* w_proj = (const float*)d_in[2];
  const float* b_proj = (const float*)d_in[3];
  // d_in[4] (mask) intentionally unused: mask recomputed analytically.
  float* out = (float*)d_out;

  const size_t per = (size_t)kB * kH * kNP * kD;   // 14,548,992 floats
  float* qb = (float*)d_ws;
  float* kb = qb + per;
  float* vb = kb + per;
  float* ob = vb + per;                            // [B, N, C] attn output

  dim3 gq(kQKV / 64, (kM + 127) / 128);            // 36 x 145
  qkv_gemm_kernel<<<gq, 128, 0, stream>>>(x, w_qkv, qb, kb, vb);

  dim3 ga(kNP / 16, kH, kB);                       // 37 x 12 x 32
  attn_kernel<<<ga, 32, 0, stream>>>(qb, kb, vb, ob);

  dim3 gp(kC / 64, (kM + 127) / 128);              // 12 x 145
  proj_gemm_kernel<<<gp, 128, 0, stream>>>(ob, w_proj, b_proj, out);
}